// MRGIN_46617575031093
// MI455X (gfx1250) — compile-verified
//
#include <hip/hip_runtime.h>
#include <hip/hip_bf16.h>

// ---------------- problem constants (match reference) ----------------
#define N_NODES 30000   // 30000 = 375 * 80 = 1875 * 16
#define N_EDGES 480000
#define N_REL   4
#define N_BATCH 16
#define F_IN    64
#define HID     256
#define LSTM_D  512
#define NCLASS  8

typedef __attribute__((ext_vector_type(2))) float v2f;
typedef __attribute__((ext_vector_type(8))) float v8f;

#if defined(__AMDGCN__) && __has_builtin(__builtin_amdgcn_wmma_f32_16x16x4_f32)
  #define GEMM_F32_WMMA 1
#endif

// ---------------------------------------------------------------------
// WMMA GEMM: C[M x Nc] = A[M x K] * op(W) (+bias) (optional relu), fp32.
// A row-major (lda), W row-major (ldw); TRANS==1 -> use W^T (elem (k,col)
// read from W[col*ldw + k]).
// One wave per (16*MT x 16) output stripe: MT stacked 16x16 tiles share
// each B fragment. K-loop is software-pipelined: fragments for step k+4
// are fetched while the WMMAs for step k execute, so the matrix ops only
// wait on the previous iteration's loads.
// All loads index directly off the kernel-arg base pointers (size_t
// offsets) so address-space inference yields global_load, not flat_load.
// grid = (M/(16*MT), Nc/16), block = 32 (one wave).
// Fragment layouts per CDNA5 ISA 7.12.2:
//   f32 A 16x4 : VGPR0 = K {0 | 2} by lane-half, VGPR1 = K {1 | 3}
//   C/D 16x16  : VGPR v -> row (v | v+8) by lane-half, col = lane&15
// ---------------------------------------------------------------------
template <int MT, int TRANS>
__global__ void wmma_gemm_t(const float* __restrict__ A, const float* __restrict__ W,
                            const float* __restrict__ bias, float* __restrict__ C,
                            int K, int lda, int ldw, int ldc, int relu) {
  const int lane = threadIdx.x & 31;
  const int half = lane >> 4;   // selects K sub-pair {0,1} vs {2,3} / row +8
  const int r    = lane & 15;
  const int m0   = blockIdx.x * (16 * MT);
  const int n0   = blockIdx.y * 16;

  v8f acc[MT];
#pragma unroll
  for (int i = 0; i < MT; ++i) acc[i] = (v8f){0.f,0.f,0.f,0.f,0.f,0.f,0.f,0.f};

#if defined(GEMM_F32_WMMA)
  size_t aoff[MT];
#pragma unroll
  for (int i = 0; i < MT; ++i)
    aoff[i] = (size_t)(m0 + i * 16 + r) * lda + half * 2;
  size_t woff = TRANS ? ((size_t)(n0 + r) * ldw + half * 2)
                      : ((size_t)(half * 2) * ldw + (n0 + r));
  const size_t wstep = TRANS ? 4 : (size_t)4 * ldw;  // k advance for W
  const size_t wdup  = TRANS ? 1 : (size_t)ldw;      // b.x -> b.y offset

  // prologue: fetch fragments for k = 0
  v2f a_c[MT], b_c;
  b_c.x = W[woff];
  b_c.y = W[woff + wdup];
  woff += wstep;
#pragma unroll
  for (int i = 0; i < MT; ++i) {
    a_c[i].x = A[aoff[i]];
    a_c[i].y = A[aoff[i] + 1];
    aoff[i] += 4;
  }

  for (int k = 4; k < K; k += 4) {
    // prefetch k's fragments while issuing WMMAs for k-4
    v2f a_n[MT], b_n;
    b_n.x = W[woff];
    b_n.y = W[woff + wdup];
    woff += wstep;
#pragma unroll
    for (int i = 0; i < MT; ++i) {
      a_n[i].x = A[aoff[i]];
      a_n[i].y = A[aoff[i] + 1];
      aoff[i] += 4;
    }
#pragma unroll
    for (int i = 0; i < MT; ++i)
      acc[i] = __builtin_amdgcn_wmma_f32_16x16x4_f32(
          false, a_c[i], false, b_c, (short)0, acc[i], false, false);
    b_c = b_n;
#pragma unroll
    for (int i = 0; i < MT; ++i) a_c[i] = a_n[i];
  }
  // epilogue: last k-step
#pragma unroll
  for (int i = 0; i < MT; ++i)
    acc[i] = __builtin_amdgcn_wmma_f32_16x16x4_f32(
        false, a_c[i], false, b_c, (short)0, acc[i], false, false);
#else
  // Host-pass / fallback: scalar reference of the same math & layout.
  for (int i = 0; i < MT; ++i)
    for (int v = 0; v < 8; ++v) {
      float s = 0.f;
      const int row = m0 + i * 16 + v + half * 8, col = n0 + r;
      for (int k = 0; k < K; ++k)
        s += A[(size_t)row * lda + k] *
             (TRANS ? W[(size_t)col * ldw + k] : W[(size_t)k * ldw + col]);
      acc[i][v] = s;
    }
#endif

#pragma unroll
  for (int i = 0; i < MT; ++i) {
#pragma unroll
    for (int v = 0; v < 8; ++v) {
      const int row = m0 + i * 16 + v + half * 8;
      const int col = n0 + r;
      float x = acc[i][v];
      if (bias) x += bias[col];
      if (relu) x = fmaxf(x, 0.f);
      C[(size_t)row * ldc + col] = x;
    }
  }
}

// ---------------- small helper kernels ----------------
__global__ void zero_f32(float* __restrict__ p, int n) {
  int i = blockIdx.x * blockDim.x + threadIdx.x;
  if (i < n) p[i] = 0.f;
}

// cnt[n, r] = number of edges with dst==n and etype==r (layer-invariant)
__global__ void count_edges(const int* __restrict__ dst, const int* __restrict__ et,
                            float* __restrict__ cnt) {
  int e = blockIdx.x * blockDim.x + threadIdx.x;
  if (e < N_EDGES) atomicAdd(&cnt[(size_t)dst[e] * N_REL + et[e]], 1.0f);
}

// per-edge mean normalizer: norm[e] = 1 / max(cnt[dst[e], et[e]], 1)
__global__ void edge_norm(const int* __restrict__ dst, const int* __restrict__ et,
                          const float* __restrict__ cnt, float* __restrict__ norm) {
  int e = blockIdx.x * blockDim.x + threadIdx.x;
  if (e < N_EDGES)
    norm[e] = 1.0f / fmaxf(cnt[(size_t)dst[e] * N_REL + et[e]], 1.0f);
}

// acc[dst] += lin_r[src] * norm[e]  for edges of relation `rel`
// 64 threads per edge, float4 per thread; lin_r/acc are 30 MB -> L2-resident.
__global__ void scatter_rel(const float* __restrict__ lin, const int* __restrict__ src,
                            const int* __restrict__ dst, const int* __restrict__ et,
                            const float* __restrict__ norm, float* __restrict__ acc,
                            int rel) {
  int tid = blockIdx.x * blockDim.x + threadIdx.x;
  int e = tid >> 6;
  int c = (tid & 63) << 2;
  if (e >= N_EDGES) return;
  if (et[e] != rel) return;
  float nrm = norm[e];
  const float4 v = *(const float4*)(lin + (size_t)src[e] * HID + c);
  float* ap = acc + (size_t)dst[e] * HID + c;
  atomicAdd(ap + 0, v.x * nrm);
  atomicAdd(ap + 1, v.y * nrm);
  atomicAdd(ap + 2, v.z * nrm);
  atomicAdd(ap + 3, v.w * nrm);
}

// h = relu(acc + bias); pooled[batch[n], poolOff + c] += h  (global_add_pool)
__global__ void combine_relu_pool(const float* __restrict__ acc, const float* __restrict__ bias,
                                  const int* __restrict__ batch, float* __restrict__ h,
                                  float* __restrict__ pooled, int poolOff) {
  int tid = blockIdx.x * blockDim.x + threadIdx.x;
  int n = tid >> 6;
  int c = (tid & 63) << 2;
  if (n >= N_NODES) return;
  const float* ap = acc + (size_t)n * HID + c;
  float4 v;
  v.x = fmaxf(ap[0] + bias[c + 0], 0.f);
  v.y = fmaxf(ap[1] + bias[c + 1], 0.f);
  v.z = fmaxf(ap[2] + bias[c + 2], 0.f);
  v.w = fmaxf(ap[3] + bias[c + 3], 0.f);
  *(float4*)(h + (size_t)n * HID + c) = v;
  float* pp = pooled + (size_t)batch[n] * (3 * HID) + poolOff + c;
  atomicAdd(pp + 0, v.x);
  atomicAdd(pp + 1, v.y);
  atomicAdd(pp + 2, v.z);
  atomicAdd(pp + 3, v.w);
}

// Sequential 16-step LSTM; gx = precomputed (seq @ Wih^T); recurrence
// h @ Whh^T here with h in LDS. One block of LSTM_D threads; then fc2 +
// log_softmax -> out[8].
__global__ void lstm_head(const float* __restrict__ gx, const float* __restrict__ Whh,
                          const float* __restrict__ bih, const float* __restrict__ bhh,
                          const float* __restrict__ fc2_w, const float* __restrict__ fc2_b,
                          float* __restrict__ out) {
  __shared__ float hs[LSTM_D];
  __shared__ float logits[NCLASS];
  const int j = threadIdx.x;
  float c = 0.f;
  hs[j] = 0.f;
  __syncthreads();
  for (int t = 0; t < N_BATCH; ++t) {
    const float* gt = gx + (size_t)t * 4 * LSTM_D;
    float gi = gt[j]              + bih[j]              + bhh[j];
    float gf = gt[LSTM_D + j]     + bih[LSTM_D + j]     + bhh[LSTM_D + j];
    float gg = gt[2 * LSTM_D + j] + bih[2 * LSTM_D + j] + bhh[2 * LSTM_D + j];
    float go = gt[3 * LSTM_D + j] + bih[3 * LSTM_D + j] + bhh[3 * LSTM_D + j];
    const float* w0 = Whh + (size_t)j * LSTM_D;
    const float* w1 = Whh + (size_t)(LSTM_D + j) * LSTM_D;
    const float* w2 = Whh + (size_t)(2 * LSTM_D + j) * LSTM_D;
    const float* w3 = Whh + (size_t)(3 * LSTM_D + j) * LSTM_D;
    for (int k = 0; k < LSTM_D; ++k) {
      float hk = hs[k];
      gi += w0[k] * hk; gf += w1[k] * hk; gg += w2[k] * hk; go += w3[k] * hk;
    }
    float si = 1.f / (1.f + __expf(-gi));
    float sf = 1.f / (1.f + __expf(-gf));
    float so = 1.f / (1.f + __expf(-go));
    c = sf * c + si * tanhf(gg);
    float hn = so * tanhf(c);
    __syncthreads();
    hs[j] = hn;
    __syncthreads();
  }
  if (j < NCLASS) {
    float s = fc2_b[j];
    for (int k = 0; k < LSTM_D; ++k) s += hs[k] * fc2_w[k * NCLASS + j];
    logits[j] = s;
  }
  __syncthreads();
  if (j == 0) {
    float m = logits[0];
    for (int i = 1; i < NCLASS; ++i) m = fmaxf(m, logits[i]);
    float sum = 0.f;
    for (int i = 0; i < NCLASS; ++i) sum += __expf(logits[i] - m);
    float lse = m + __logf(sum);
    for (int i = 0; i < NCLASS; ++i) out[i] = logits[i] - lse;
  }
}

// ---------------------------------------------------------------------
extern "C" void kernel_launch(void* const* d_in, const int* in_sizes, int n_in,
                              void* d_out, int out_size, void* d_ws, size_t ws_size,
                              hipStream_t stream) {
  (void)in_sizes; (void)n_in; (void)out_size; (void)ws_size;

  const float* x      = (const float*)d_in[0];
  const int*   eidx   = (const int*)d_in[1];
  const int*   src    = eidx;                // edge_index[0]
  const int*   dst    = eidx + N_EDGES;      // edge_index[1]
  const int*   etype  = (const int*)d_in[2];
  const int*   batch  = (const int*)d_in[3];
  const float* Wrel0  = (const float*)d_in[4];
  const float* Wroot0 = (const float*)d_in[5];
  const float* b0     = (const float*)d_in[6];
  const float* Wrel12 = (const float*)d_in[7];
  const float* Wroot12= (const float*)d_in[8];
  const float* b12    = (const float*)d_in[9];
  const float* fc1_w  = (const float*)d_in[10];
  const float* fc1_b  = (const float*)d_in[11];
  const float* Wih    = (const float*)d_in[12];
  const float* Whh    = (const float*)d_in[13];
  const float* bih    = (const float*)d_in[14];
  const float* bhh    = (const float*)d_in[15];
  const float* fc2_w  = (const float*)d_in[16];
  const float* fc2_b  = (const float*)d_in[17];
  float* out = (float*)d_out;

  // workspace layout (floats): ~95 MB total
  float* ws     = (float*)d_ws;
  float* h      = ws;                               // N*HID   layer activations
  float* acc    = h    + (size_t)N_NODES * HID;     // N*HID   root + aggregated msgs
  float* linr   = acc  + (size_t)N_NODES * HID;     // N*HID   per-relation transform
  float* cnt    = linr + (size_t)N_NODES * HID;     // N*R
  float* pooled = cnt  + (size_t)N_NODES * N_REL;   // B * 3*HID
  float* enorm  = pooled + N_BATCH * 3 * HID;       // E       per-edge 1/deg
  float* seq    = enorm  + N_EDGES;                 // B * LSTM_D
  float* gx     = seq    + N_BATCH * LSTM_D;        // B * 4*LSTM_D

  // zero cnt + pooled (contiguous); count edges; precompute per-edge norm
  const int nz = N_NODES * N_REL + N_BATCH * 3 * HID;
  zero_f32<<<(nz + 255) / 256, 256, 0, stream>>>(cnt, nz);
  count_edges<<<N_EDGES / 256, 256, 0, stream>>>(dst, etype, cnt);
  edge_norm<<<N_EDGES / 256, 256, 0, stream>>>(dst, etype, cnt, enorm);

  const dim3 gN(N_NODES / 80, HID / 16);   // MT = 5 -> 375 x 16 waves
  const int scatterBlocks = (N_EDGES * 64) / 256;
  const int combineBlocks = (N_NODES * 64) / 256;

  // ---- layer 0 (input x, K = F_IN) ----
  wmma_gemm_t<5, 0><<<gN, 32, 0, stream>>>(x, Wroot0, nullptr, acc,
                                           F_IN, F_IN, HID, HID, 0);
  for (int r = 0; r < N_REL; ++r) {
    wmma_gemm_t<5, 0><<<gN, 32, 0, stream>>>(x, Wrel0 + (size_t)r * F_IN * HID,
                                             nullptr, linr, F_IN, F_IN, HID, HID, 0);
    scatter_rel<<<scatterBlocks, 256, 0, stream>>>(linr, src, dst, etype, enorm, acc, r);
  }
  combine_relu_pool<<<combineBlocks, 256, 0, stream>>>(acc, b0, batch, h, pooled, 0);

  // ---- layers 1, 2 (input h, K = HID) ----
  for (int L = 0; L < 2; ++L) {
    const float* Wroot = Wroot12 + (size_t)L * HID * HID;
    const float* bL    = b12 + (size_t)L * HID;
    wmma_gemm_t<5, 0><<<gN, 32, 0, stream>>>(h, Wroot, nullptr, acc,
                                             HID, HID, HID, HID, 0);
    for (int r = 0; r < N_REL; ++r) {
      const float* Wr = Wrel12 + (size_t)(L * N_REL + r) * HID * HID;
      wmma_gemm_t<5, 0><<<gN, 32, 0, stream>>>(h, Wr, nullptr, linr,
                                               HID, HID, HID, HID, 0);
      scatter_rel<<<scatterBlocks, 256, 0, stream>>>(linr, src, dst, etype, enorm, acc, r);
    }
    combine_relu_pool<<<combineBlocks, 256, 0, stream>>>(acc, bL, batch, h, pooled,
                                                         (L + 1) * HID);
  }

  // ---- head: fc1 (bias+relu), then input-side LSTM gates as a GEMM ----
  wmma_gemm_t<1, 0><<<dim3(1, LSTM_D / 16), 32, 0, stream>>>(
      pooled, fc1_w, fc1_b, seq, 3 * HID, 3 * HID, LSTM_D, LSTM_D, 1);
  // gx = seq @ Wih^T   (Wih is [4*LSTM_D, LSTM_D] row-major -> TRANS)
  wmma_gemm_t<1, 1><<<dim3(1, (4 * LSTM_D) / 16), 32, 0, stream>>>(
      seq, Wih, nullptr, gx, LSTM_D, LSTM_D, LSTM_D, 4 * LSTM_D, 0);

  // ---- recurrent LSTM + fc2 + log_softmax ----
  lstm_head<<<1, LSTM_D, 0, stream>>>(gx, Whh, bih, bhh, fc2_w, fc2_b, out);
}